// GAT_1709396984304
// MI455X (gfx1250) — compile-verified
//
#include <hip/hip_runtime.h>
#include <math.h>

typedef __attribute__((ext_vector_type(2))) float v2f;
typedef __attribute__((ext_vector_type(8))) float v8f;

#define NEG_SLOPE 0.2f

// ---------- helpers: order-preserving float<->uint map for atomicMax on floats ----------
__device__ __forceinline__ unsigned f2o(float f) {
    unsigned u = __float_as_uint(f);
    return (u & 0x80000000u) ? ~u : (u | 0x80000000u);
}
__device__ __forceinline__ float o2f(unsigned u) {
    return (u & 0x80000000u) ? __uint_as_float(u & 0x7fffffffu) : __uint_as_float(~u);
}
#define ORD_NEG_INF 0x007FFFFFu   // f2o(-inf)

// ---------- generic zero ----------
__global__ void zeroF(float* __restrict__ p, int n) {
    int i = blockIdx.x * blockDim.x + threadIdx.x;
    if (i < n) p[i] = 0.0f;
}

// ---------- self-loop mean attr: segment sum + count over dst ----------
__global__ void edgeMeanK(const int* __restrict__ ei, const float* __restrict__ ea,
                          float* __restrict__ s, float* __restrict__ c, int E) {
    int e = blockIdx.x * blockDim.x + threadIdx.x;
    if (e >= E) return;
    int dst = ei[E + e];
    atomicAdd(&s[dst], ea[e]);
    atomicAdd(&c[dst], 1.0f);
}
__global__ void loopAttrK(float* __restrict__ s, const float* __restrict__ c, int N) {
    int i = blockIdx.x * blockDim.x + threadIdx.x;
    if (i < N) s[i] = s[i] / fmaxf(c[i], 1.0f);   // in place: s becomes loop_attr
}

// ---------- k = dot(W_edge[0,:], att_edge) for both layers (edge_dim == 1) ----------
__global__ void kdotK(const float* __restrict__ we1, const float* __restrict__ ae1,
                      const float* __restrict__ we2, const float* __restrict__ ae2,
                      float* __restrict__ kscal) {
    int t = threadIdx.x;
    if (t == 0) { float s = 0.f; for (int j = 0; j < 64; ++j) s += we1[j] * ae1[j]; kscal[0] = s; }
    if (t == 1) { float s = 0.f; for (int j = 0; j < 32; ++j) s += we2[j] * ae2[j]; kscal[1] = s; }
}

// ---------- fp32 GEMM via V_WMMA_F32_16X16X4_F32 ----------
// C[M, NT*16] = A[M, KDIM] @ B[KDIM, NT*16].   One wave per 16-row stripe:
// A fragment loaded once per K-step and reused across NT output tiles.
// Compile-time KDIM/NCOL => all loads are base + immediate offset.
template <int KDIM, int NT>
__global__ void gemmWmmaK(const float* __restrict__ A, const float* __restrict__ B,
                          float* __restrict__ C, int mtiles) {
    constexpr int NCOL = NT * 16;
    int wave = (blockIdx.x * blockDim.x + threadIdx.x) >> 5;
    int lane = threadIdx.x & 31;
    if (wave >= mtiles) return;                  // wave-uniform; EXEC stays all-ones

    int half = lane >> 4;                        // 0: lanes 0-15, 1: lanes 16-31
    int idx  = lane & 15;                        // A: row M, B: col N (within tile)
    int kk   = half << 1;                        // K offset within 4-step: 0 or 2

    const float* arow  = A + (wave * 16 + idx) * KDIM + kk;
    const float* bbase = B + kk * NCOL + idx;

    v8f acc[NT];
#pragma unroll
    for (int t = 0; t < NT; ++t) acc[t] = (v8f){};

#pragma unroll
    for (int k0 = 0; k0 < KDIM; k0 += 4) {
        v2f a;
        a.x = arow[k0];
        a.y = arow[k0 + 1];
#pragma unroll
        for (int t = 0; t < NT; ++t) {
            v2f b;
            b.x = bbase[k0 * NCOL + t * 16];
            b.y = bbase[(k0 + 1) * NCOL + t * 16];
            acc[t] = __builtin_amdgcn_wmma_f32_16x16x4_f32(false, a, false, b,
                                                           (short)0, acc[t], false, false);
        }
    }
    // C/D layout: VGPR i -> row i (lanes 0-15) / row i+8 (lanes 16-31); col = lane & 15
    int rbase = wave * 16 + (half << 3);
#pragma unroll
    for (int t = 0; t < NT; ++t) {
        float* cp = C + rbase * NCOL + t * 16 + idx;
#pragma unroll
        for (int i = 0; i < 8; ++i) cp[i * NCOL] = acc[t][i];
    }
}

// ---------- per-node attention dots: a_src[i] = h[i,:].att_src ; a_dst likewise ----------
__global__ void attdotK(const float* __restrict__ h, const float* __restrict__ as,
                        const float* __restrict__ ad, float* __restrict__ osrc,
                        float* __restrict__ odst, int N, int d4) {
    int i = blockIdx.x * blockDim.x + threadIdx.x;
    if (i >= N) return;
    const float4* hp = (const float4*)(h + (size_t)i * (d4 * 4));
    const float4* ap = (const float4*)as;
    const float4* bp = (const float4*)ad;
    float s1 = 0.f, s2 = 0.f;
    for (int j = 0; j < d4; ++j) {
        float4 v = hp[j], a = ap[j], b = bp[j];
        s1 += v.x * a.x + v.y * a.y + v.z * a.z + v.w * a.w;
        s2 += v.x * b.x + v.y * b.y + v.z * b.z + v.w * b.w;
    }
    osrc[i] = s1; odst[i] = s2;
}

// ---------- softmax state init ----------
__global__ void initMDK(unsigned* __restrict__ m, float* __restrict__ den, int N) {
    int i = blockIdx.x * blockDim.x + threadIdx.x;
    if (i < N) { m[i] = ORD_NEG_INF; den[i] = 0.0f; }
}

// ---------- per-edge alpha + segment max (edges [0,E) real, [E,ET) self-loops) ----------
__global__ void alphaK(const int* __restrict__ ei, const float* __restrict__ ea,
                       const float* __restrict__ la, const float* __restrict__ asrc,
                       const float* __restrict__ adst, const float* __restrict__ kscal,
                       int ki, float* __restrict__ alpha, unsigned* __restrict__ m,
                       int E, int ET) {
    int e = blockIdx.x * blockDim.x + threadIdx.x;
    if (e >= ET) return;
    int s, dd; float attr;
    if (e < E) { s = ei[e]; dd = ei[E + e]; attr = ea[e]; }
    else       { s = dd = e - E; attr = la[e - E]; }
    float al = asrc[s] + adst[dd] + attr * kscal[ki];
    al = (al > 0.0f) ? al : NEG_SLOPE * al;
    alpha[e] = al;
    atomicMax(&m[dd], f2o(al));
}

// ---------- exp(alpha - m[dst]) + segment sum ----------
__global__ void expSumK(const int* __restrict__ ei, const float* __restrict__ alpha,
                        const unsigned* __restrict__ m, float* __restrict__ den,
                        float* __restrict__ ex, int E, int ET) {
    int e = blockIdx.x * blockDim.x + threadIdx.x;
    if (e >= ET) return;
    int dd = (e < E) ? ei[E + e] : (e - E);
    float v = expf(alpha[e] - o2f(m[dd]));
    ex[e] = v;
    atomicAdd(&den[dd], v);
}

// ---------- weighted scatter-add: out[dst,:] += (ex/den[dst]) * h[src,:] ----------
// d threads per edge (d = 64 or 32, power of two; shift = log2(d))
__global__ void aggK(const int* __restrict__ ei, const float* __restrict__ ex,
                     const float* __restrict__ den, const float* __restrict__ h,
                     float* __restrict__ accum, int shift, int E, int ET) {
    int idx = blockIdx.x * blockDim.x + threadIdx.x;
    int e = idx >> shift;
    if (e >= ET) return;
    int j = idx & ((1 << shift) - 1);
    int s, dd;
    if (e < E) { s = ei[e]; dd = ei[E + e]; }
    else       { s = dd = e - E; }
    float w = ex[e] / den[dd];
    atomicAdd(&accum[(dd << shift) + j], w * h[(s << shift) + j]);
}

// ---------- epilogues ----------
__global__ void biasReluK(float* __restrict__ p, const float* __restrict__ b, int n, int mask) {
    int i = blockIdx.x * blockDim.x + threadIdx.x;
    if (i < n) p[i] = fmaxf(p[i] + b[i & mask], 0.0f);
}
__global__ void biasK(float* __restrict__ p, const float* __restrict__ b, int n, int mask) {
    int i = blockIdx.x * blockDim.x + threadIdx.x;
    if (i < n) p[i] = p[i] + b[i & mask];
}

static inline int gr(long long n, int blk) { return (int)((n + blk - 1) / blk); }

extern "C" void kernel_launch(void* const* d_in, const int* in_sizes, int n_in,
                              void* d_out, int out_size, void* d_ws, size_t ws_size,
                              hipStream_t stream) {
    const float* x    = (const float*)d_in[0];
    const int*   ei   = (const int*)d_in[1];
    const float* ea   = (const float*)d_in[2];
    const float* W1   = (const float*)d_in[3];
    const float* as1  = (const float*)d_in[4];
    const float* ad1  = (const float*)d_in[5];
    const float* we1  = (const float*)d_in[6];
    const float* ae1  = (const float*)d_in[7];
    const float* b1   = (const float*)d_in[8];
    const float* W2   = (const float*)d_in[9];
    const float* as2  = (const float*)d_in[10];
    const float* ad2  = (const float*)d_in[11];
    const float* we2  = (const float*)d_in[12];
    const float* ae2  = (const float*)d_in[13];
    const float* b2   = (const float*)d_in[14];
    float* out = (float*)d_out;

    const int N  = in_sizes[0] / 128;   // 50000
    const int E  = in_sizes[1] / 2;     // 800000
    const int ET = E + N;

    // workspace layout (floats)
    float* ws = (float*)d_ws;
    size_t o = 0;
    float*    la    = ws + o; o += N;              // segment sum -> loop_attr (in place)
    float*    cnt   = ws + o; o += N;
    float*    kscal = ws + o; o += 4;
    float*    h     = ws + o; o += (size_t)N * 64; // h1, reused for h2 (N*32)
    float*    asrc  = ws + o; o += N;
    float*    adst  = ws + o; o += N;
    unsigned* mbuf  = (unsigned*)(ws + o); o += N;
    float*    den   = ws + o; o += N;
    float*    alpha = ws + o; o += ET;
    float*    ex    = ws + o; o += ET;
    float*    hagg  = ws + o; o += (size_t)N * 64; // layer-1 output / layer-2 GEMM input

    const int B = 256;
    const int mtiles = N / 16;                     // 3125 (exact)

    // ----- self-loop mean edge attr -----
    zeroF<<<gr(2 * (long long)N, B), B, 0, stream>>>(la, 2 * N);     // la & cnt contiguous
    edgeMeanK<<<gr(E, B), B, 0, stream>>>(ei, ea, la, cnt, E);
    loopAttrK<<<gr(N, B), B, 0, stream>>>(la, cnt, N);
    kdotK<<<1, 64, 0, stream>>>(we1, ae1, we2, ae2, kscal);

    // ===== layer 1: 128 -> 64 =====
    gemmWmmaK<128, 4><<<gr(mtiles, 8), 8 * 32, 0, stream>>>(x, W1, h, mtiles);
    attdotK<<<gr(N, B), B, 0, stream>>>(h, as1, ad1, asrc, adst, N, 16);
    initMDK<<<gr(N, B), B, 0, stream>>>(mbuf, den, N);
    zeroF<<<gr((long long)N * 64, B), B, 0, stream>>>(hagg, N * 64);
    alphaK<<<gr(ET, B), B, 0, stream>>>(ei, ea, la, asrc, adst, kscal, 0, alpha, mbuf, E, ET);
    expSumK<<<gr(ET, B), B, 0, stream>>>(ei, alpha, mbuf, den, ex, E, ET);
    aggK<<<gr((long long)ET * 64, B), B, 0, stream>>>(ei, ex, den, h, hagg, 6, E, ET);
    biasReluK<<<gr((long long)N * 64, B), B, 0, stream>>>(hagg, b1, N * 64, 63);

    // ===== layer 2: 64 -> 32 =====
    gemmWmmaK<64, 2><<<gr(mtiles, 8), 8 * 32, 0, stream>>>(hagg, W2, h, mtiles);
    attdotK<<<gr(N, B), B, 0, stream>>>(h, as2, ad2, asrc, adst, N, 8);
    initMDK<<<gr(N, B), B, 0, stream>>>(mbuf, den, N);
    zeroF<<<gr((long long)N * 32, B), B, 0, stream>>>(out, N * 32);
    alphaK<<<gr(ET, B), B, 0, stream>>>(ei, ea, la, asrc, adst, kscal, 1, alpha, mbuf, E, ET);
    expSumK<<<gr(ET, B), B, 0, stream>>>(ei, alpha, mbuf, den, ex, E, ET);
    aggK<<<gr((long long)ET * 32, B), B, 0, stream>>>(ei, ex, den, h, out, 5, E, ET);
    biasK<<<gr((long long)N * 32, B), B, 0, stream>>>(out, b2, N * 32, 31);
}